// PEGenerator_81990925680868
// MI455X (gfx1250) — compile-verified
//
#include <hip/hip_runtime.h>

// ---------------------------------------------------------------------------
// PointNet++-style ball query + tiny MLP + BN + max-pool, MI455X (gfx1250).
// b=2, n=8192, c=64, k=32.  All fp32 (compute-light, memory-light problem:
// 4MB out, <1GFLOP total -> instruction efficiency dominates, not peak math).
// WMMA f32 16x16x4 used for the per-point (32x3)@(3x64) matmul + max pool.
// ---------------------------------------------------------------------------

typedef __attribute__((ext_vector_type(2))) float v2f;
typedef __attribute__((ext_vector_type(8))) float v8f;

#define KNN   32      // k (matches reference setup)
#define CH    1024    // candidates staged in LDS per chunk
#define QPB   8       // query points per 256-thread block (one wave each)
#define CCH   64      // output channels

// ---------------------------------------------------------------------------
// Kernel 1: ball query (first-KNN ascending-index compaction), h = rel @ w1^T,
// write h_pre (incl. pad slots = first-neighbor replica), accumulate BN
// partial sums per block (deterministic, no float atomics).
// ---------------------------------------------------------------------------
__global__ __launch_bounds__(256) void bq_h_kernel(
    const float* __restrict__ pos, const float* __restrict__ w1,
    const float* __restrict__ radius_p,
    float* __restrict__ hpre, float* __restrict__ partials, int n_per_b)
{
  __shared__ float cx[CH], cy[CH], cz[CH], cs[CH];
  __shared__ float redbuf[QPB * 6];

  const int tid  = threadIdx.x;
  const int lane = tid & 31;
  const int wave = tid >> 5;
  const int q    = blockIdx.x * QPB + wave;   // global point id in [0, b*n)
  const int bidx = q / n_per_b;               // uniform per block (n_per_b % QPB == 0)
  const int base = bidx * n_per_b;

  const float r  = radius_p[0];
  const float r2 = r * r;

  const float qx = pos[q * 3 + 0], qy = pos[q * 3 + 1], qz = pos[q * 3 + 2];
  const float sqq = qx * qx + qy * qy + qz * qz;

  const float w00 = w1[0], w01 = w1[1], w02 = w1[2];
  const float w10 = w1[3], w11 = w1[4], w12 = w1[5];
  const float w20 = w1[6], w21 = w1[7], w22 = w1[8];

  int  cnt  = 0;           // uniform across the wave (ballot-derived)
  bool done = false;
  float fh0 = 0.f, fh1 = 0.f, fh2 = 0.f;                 // first-neighbor h
  float s0 = 0.f, s1 = 0.f, s2 = 0.f;                    // BN sum
  float p0 = 0.f, p1 = 0.f, p2 = 0.f;                    // BN sum of squares

  for (int chunk = 0; chunk < n_per_b; chunk += CH) {
    __syncthreads();
    for (int t = tid; t < CH; t += 256) {
      const int g = (base + chunk + t) * 3;
      const float px = pos[g], py = pos[g + 1], pz = pos[g + 2];
      cx[t] = px; cy[t] = py; cz[t] = pz;
      cs[t] = px * px + py * py + pz * pz;
    }
    __syncthreads();
    if (done) continue;   // done waves still participate in barriers above

    for (int t0 = 0; t0 < CH; t0 += 32) {
      const int t = t0 + lane;
      const float px = cx[t], py = cy[t], pz = cz[t];
      const float dot = qx * px + qy * py + qz * pz;
      const float d2  = (sqq + cs[t]) - 2.0f * dot;   // same Gram form as reference
      const bool pass = d2 < r2;
      const unsigned m = (unsigned)__ballot(pass);
      if (m) {
        const float rx = px - qx, ry = py - qy, rz = pz - qz;
        const float h0 = w00 * rx + w01 * ry + w02 * rz;
        const float h1 = w10 * rx + w11 * ry + w12 * rz;
        const float h2 = w20 * rx + w21 * ry + w22 * rz;
        const int rank = __popc(m & ((1u << lane) - 1u));
        const int slot = cnt + rank;
        if (pass && slot < KNN) {
          const size_t hb = ((size_t)q * KNN + slot) * 3;
          hpre[hb + 0] = h0; hpre[hb + 1] = h1; hpre[hb + 2] = h2;
          s0 += h0; s1 += h1; s2 += h2;
          p0 += h0 * h0; p1 += h1 * h1; p2 += h2 * h2;
        }
        if (cnt == 0) {
          const int src = __builtin_ctz(m);   // lowest-index passer => slot 0
          fh0 = __shfl(h0, src, 32);
          fh1 = __shfl(h1, src, 32);
          fh2 = __shfl(h2, src, 32);
        }
        cnt += __popc(m);
        if (cnt >= KNN) { done = true; break; }
      }
    }
  }

  // Pad slots with first-neighbor h (counted in BN stats, as in reference).
  const int found = cnt < KNN ? cnt : KNN;
  const int rem   = KNN - found;
  if (rem > 0) {
    const int slot = found + lane;
    if (slot < KNN) {
      const size_t hb = ((size_t)q * KNN + slot) * 3;
      hpre[hb + 0] = fh0; hpre[hb + 1] = fh1; hpre[hb + 2] = fh2;
    }
    if (lane == 0) {
      const float fr = (float)rem;
      s0 += fr * fh0; s1 += fr * fh1; s2 += fr * fh2;
      p0 += fr * fh0 * fh0; p1 += fr * fh1 * fh1; p2 += fr * fh2 * fh2;
    }
  }

  // Wave reduce 6 accumulators, then cross-wave via LDS -> per-block partial.
  for (int off = 16; off > 0; off >>= 1) {
    s0 += __shfl_xor(s0, off, 32); s1 += __shfl_xor(s1, off, 32); s2 += __shfl_xor(s2, off, 32);
    p0 += __shfl_xor(p0, off, 32); p1 += __shfl_xor(p1, off, 32); p2 += __shfl_xor(p2, off, 32);
  }
  if (lane == 0) {
    redbuf[wave * 6 + 0] = s0; redbuf[wave * 6 + 1] = s1; redbuf[wave * 6 + 2] = s2;
    redbuf[wave * 6 + 3] = p0; redbuf[wave * 6 + 4] = p1; redbuf[wave * 6 + 5] = p2;
  }
  __syncthreads();
  if (tid < 6) {
    float tot = 0.f;
    for (int w = 0; w < QPB; ++w) tot += redbuf[w * 6 + tid];
    partials[(size_t)blockIdx.x * 8 + tid] = tot;
  }
}

// ---------------------------------------------------------------------------
// Kernel 2: fold per-block partials -> mean/var -> scale = gamma*rsqrt(var+eps),
// shift = beta - mean*scale.  Writes stats[0..5].
// ---------------------------------------------------------------------------
__global__ __launch_bounds__(256) void stats_kernel(
    const float* __restrict__ partials, int nparts,
    const float* __restrict__ gamma, const float* __restrict__ beta,
    float* __restrict__ stats, float inv_count)
{
  __shared__ float red[256];
  float loc[6] = {0.f, 0.f, 0.f, 0.f, 0.f, 0.f};
  for (int r = threadIdx.x; r < nparts; r += 256)
    for (int c = 0; c < 6; ++c) loc[c] += partials[(size_t)r * 8 + c];

  float tot[6];
  for (int c = 0; c < 6; ++c) {
    __syncthreads();
    red[threadIdx.x] = loc[c];
    __syncthreads();
    for (int off = 128; off > 0; off >>= 1) {
      if (threadIdx.x < off) red[threadIdx.x] += red[threadIdx.x + off];
      __syncthreads();
    }
    tot[c] = red[0];
  }
  if (threadIdx.x == 0) {
    for (int d = 0; d < 3; ++d) {
      const float mean = tot[d] * inv_count;
      const float var  = tot[3 + d] * inv_count - mean * mean;
      const float inv  = rsqrtf(var + 1e-5f);
      const float sc   = inv * gamma[d];
      stats[d]     = sc;
      stats[3 + d] = beta[d] - mean * sc;
    }
  }
}

// ---------------------------------------------------------------------------
// Kernel 3: per point, normalize+ReLU h (32x3), pe = h @ w2^T via
// v_wmma_f32_16x16x4_f32 (K=3 padded to 4), max over the 32 neighbors via
// 16 in-lane v_max on the D fragment + one xor-16 shuffle, + b2 + x.
// One wave (32-thread block) per point; b2/max commute so b2 added post-max.
// A layout (16x4 f32): lanes 0-15 hold (K0,K1), lanes 16-31 hold (K2,K3);
// B mirrors with N in place of M. D: lane l = col l%16, VGPR v = row v+8*(l/16).
// All fragment-element loads are issued unconditionally (addresses in-bounds
// for every lane) and zeroed with a select afterwards, so no exec-mask
// branches appear between the WMMAs.
// ---------------------------------------------------------------------------
__global__ __launch_bounds__(32) void pe_kernel(
    const float* __restrict__ hpre, const float* __restrict__ stats,
    const float* __restrict__ w2, const float* __restrict__ b2,
    const float* __restrict__ x, float* __restrict__ out)
{
  __shared__ float shh[KNN * 3];
  const int lane = threadIdx.x;
  const int p    = blockIdx.x;

  const float sc0 = stats[0], sc1 = stats[1], sc2 = stats[2];
  const float of0 = stats[3], of1 = stats[4], of2 = stats[5];

  const size_t hb = (size_t)p * KNN * 3 + (size_t)lane * 3;
  float h0 = hpre[hb + 0], h1 = hpre[hb + 1], h2 = hpre[hb + 2];
  h0 = fmaxf(h0 * sc0 + of0, 0.f);
  h1 = fmaxf(h1 * sc1 + of1, 0.f);
  h2 = fmaxf(h2 * sc2 + of2, 0.f);
  shh[lane * 3 + 0] = h0; shh[lane * 3 + 1] = h1; shh[lane * 3 + 2] = h2;
  __syncthreads();

  const int nl = lane & 15;
  const int hi = lane >> 4;
  const float kmask = hi ? 0.f : 1.f;   // zeroes the K=3 pad column

  // A fragments (unconditional LDS loads, then mask the pad element).
  v2f a0, a1;
  {
    const float x0 = shh[nl * 3 + 2 * hi];
    const float y0 = shh[nl * 3 + 1];
    const float x1 = shh[(16 + nl) * 3 + 2 * hi];
    const float y1 = shh[(16 + nl) * 3 + 1];
    a0.x = x0; a0.y = y0 * kmask;
    a1.x = x1; a1.y = y1 * kmask;
  }

  // B fragments for all 4 channel tiles, hoisted out of the WMMA loop.
  v2f bf[4];
#pragma unroll
  for (int nb = 0; nb < 4; ++nb) {
    const int e = nb * 16 + nl;
    const float bx = w2[e * 3 + 2 * hi];
    const float by = w2[e * 3 + 1];
    bf[nb].x = bx;
    bf[nb].y = by * kmask;
  }

  // Residual + bias terms, hoisted (read by lanes 0-15 only is fine, but
  // issue for all lanes to keep the loop branch-free; addresses in-bounds).
  const size_t ob = (size_t)p * CCH;
  float xb[4];
#pragma unroll
  for (int nb = 0; nb < 4; ++nb) {
    const int e = nb * 16 + nl;
    xb[nb] = x[ob + e] + b2[e];
  }

#pragma unroll
  for (int nb = 0; nb < 4; ++nb) {
    v8f c = {};
    v8f d0 = __builtin_amdgcn_wmma_f32_16x16x4_f32(
        false, a0, false, bf[nb], (short)0, c, false, false);
    v8f d1 = __builtin_amdgcn_wmma_f32_16x16x4_f32(
        false, a1, false, bf[nb], (short)0, c, false, false);

    float m = d0[0];
#pragma unroll
    for (int i = 1; i < 8; ++i) m = fmaxf(m, d0[i]);
#pragma unroll
    for (int i = 0; i < 8; ++i) m = fmaxf(m, d1[i]);
    m = fmaxf(m, __shfl_xor(m, 16, 32));   // combine rows 0-7 with 8-15 halves

    if (lane < 16) out[ob + nb * 16 + nl] = xb[nb] + m;
  }
}

// ---------------------------------------------------------------------------
// Host launcher. d_in order: pos, x, w1, gamma, beta, w2, b2, radius, k.
// d_ws layout (floats): [0..5] stats | [16 ..) per-block partials (8 each) |
// [32768 ..) h_pre (P*KNN*3).
// ---------------------------------------------------------------------------
extern "C" void kernel_launch(void* const* d_in, const int* in_sizes, int n_in,
                              void* d_out, int out_size, void* d_ws, size_t ws_size,
                              hipStream_t stream) {
  const float* pos    = (const float*)d_in[0];
  const float* x      = (const float*)d_in[1];
  const float* w1     = (const float*)d_in[2];
  const float* gamma  = (const float*)d_in[3];
  const float* beta   = (const float*)d_in[4];
  const float* w2     = (const float*)d_in[5];
  const float* b2     = (const float*)d_in[6];
  const float* radius = (const float*)d_in[7];

  const int P = in_sizes[0] / 3;       // b * n = 16384
  const int BATCH = 2;                 // reference setup
  const int n_per_b = P / BATCH;       // 8192

  float* ws       = (float*)d_ws;
  float* stats    = ws;                // 6 floats
  float* partials = ws + 16;           // nblocks1 * 8 floats
  float* hpre     = ws + 32768;        // P * KNN * 3 floats (~6 MB)

  const int nblocks1 = P / QPB;        // 2048

  bq_h_kernel<<<nblocks1, 256, 0, stream>>>(pos, w1, radius, hpre, partials, n_per_b);
  stats_kernel<<<1, 256, 0, stream>>>(partials, nblocks1, gamma, beta, stats,
                                      1.0f / (float)((size_t)P * KNN));
  pe_kernel<<<P, 32, 0, stream>>>(hpre, stats, w2, b2, x, (float*)d_out);
}